// AminoAcidFeature_63024350101676
// MI455X (gfx1250) — compile-verified
//
#include <hip/hip_runtime.h>
#include <hip/hip_bf16.h>
#include <math.h>

#define BIGINT_F 1.0e10f
#define MAXL 128
#define MAXC 14

typedef __attribute__((ext_vector_type(2))) float v2f;
typedef __attribute__((ext_vector_type(8))) float v8f;

// ---------------------------------------------------------------------------
// Kernel 1: residue/atom embedding  h = concat(res_embed[S] + pos_e, mean_atom)
// One block per residue n, thread e handles embedding dim e (E=256) of both halves.
// ---------------------------------------------------------------------------
__global__ void embed_kernel(const int* __restrict__ S, const int* __restrict__ RP,
                             const int* __restrict__ A, const int* __restrict__ AP,
                             const float* __restrict__ res_embed,
                             const float* __restrict__ atom_embed,
                             const float* __restrict__ atom_pos_embed,
                             float* __restrict__ out, int C, int E) {
  int n = blockIdx.x;
  int e = threadIdx.x;                      // 0..E-1
  int k = e >> 1;
  float inv = powf(10000.0f, -2.0f * (float)k / (float)E);
  float ang = (float)RP[n] * inv;
  float pe  = (e & 1) ? cosf(ang) : sinf(ang);
  size_t base = (size_t)n * (size_t)(2 * E);
  out[base + e] = res_embed[S[n] * E + e] + pe;

  float acc = 0.0f, cnt = 0.0f;
  for (int c = 0; c < C; ++c) {
    int ap = AP[n * C + c];
    if (ap != 0) {
      acc += atom_embed[A[n * C + c] * E + e] + atom_pos_embed[ap * E + e];
      cnt += 1.0f;
    }
  }
  out[base + E + e] = acc / (cnt + 1e-10f);
}

// ---------------------------------------------------------------------------
// Kernel 2: min pairwise atom distance via V_WMMA_F32_16X16X4_F32.
// One wave per (b,i) row; A = -2*Xi (16x4) register-resident across the j-loop;
// C accumulator preloaded with |Xi_m|^2 + |Xj_n|^2 so one WMMA yields the full
// 16x16 squared-distance tile. The batch's X/AP block is staged in LDS (all
// waves of a block share one batch). sqrt is hoisted out of the atom-pair min
// (monotonic), so only one raw v_sqrt_f32 per (i,j).
// ---------------------------------------------------------------------------
__global__ void __launch_bounds__(128)
dist_kernel(const float* __restrict__ X, const int* __restrict__ AP,
            float* __restrict__ dist, int B, int L, int C) {
  __shared__ float sX[MAXL * MAXC * 3];
  __shared__ int   sAP[MAXL * MAXC];

  const int wavesPerBlock = blockDim.x >> 5;
  int wave0 = blockIdx.x * wavesPerBlock;
  int wave  = wave0 + (threadIdx.x >> 5);
  int lane  = threadIdx.x & 31;
  int b     = wave0 / L;                    // L % wavesPerBlock == 0 -> uniform per block
  int i     = wave - b * L;

  // Stage this batch's coordinates and pad table into LDS.
  const float* Xb  = X  + (size_t)b * L * C * 3;
  const int*   APb = AP + (size_t)b * L * C;
  for (int t = threadIdx.x; t < L * C * 3; t += blockDim.x) sX[t]  = Xb[t];
  for (int t = threadIdx.x; t < L * C;     t += blockDim.x) sAP[t] = APb[t];
  __syncthreads();

  int  mrow  = lane & 15;                   // row (M for A-half / N for B,C,D)
  bool hi    = lane >= 16;                  // upper half-wave: K=2,3 / M+=8
  int  mbase = hi ? 8 : 0;
  int  ar    = (mrow < C) ? mrow : 0;       // clamped in-bounds atom index
  float vld  = (mrow < C) ? 1.0f : 0.0f;

  // Residue i atom coords for this lane's row (from LDS).
  const float* xi = sX + (i * C + ar) * 3;
  float pix = xi[0] * vld, piy = xi[1] * vld, piz = xi[2] * vld;

  // A matrix (16x4 f32): lanes 0-15 -> K0,K1 ; lanes 16-31 -> K2,K3. Pre-scaled by -2.
  v2f areg;
  areg.x = hi ? (-2.0f * piz) : (-2.0f * pix);
  areg.y = hi ? 0.0f          : (-2.0f * piy);

  // |Xi_row|^2 per lane; gather the 8 M-values this lane's D regs cover.
  float sqv = pix * pix + piy * piy + piz * piz;
  float sqi[8];
#pragma unroll
  for (int r = 0; r < 8; ++r) sqi[r] = __shfl(sqv, mbase + r, 32);

  // Pad bitmask for residue i rows (rows >= C are pad too).
  bool padi = (mrow >= C) || (sAP[i * C + ar] == 0);
  unsigned maski = ((unsigned)__ballot(padi)) & 0xFFFFu;

#pragma unroll 4
  for (int j = 0; j < L; ++j) {
    const float* xj = sX + (j * C + ar) * 3;
    float pjx = xj[0] * vld, pjy = xj[1] * vld, pjz = xj[2] * vld;

    // B matrix (4x16): VGPR0 = row K0 (lanes 0-15) / K2 (lanes 16-31); VGPR1 = K1/K3.
    v2f breg;
    breg.x = hi ? pjz : pjx;
    breg.y = hi ? 0.0f : pjy;

    float sqj  = pjx * pjx + pjy * pjy + pjz * pjz;       // this lane's N = mrow
    bool  padj = (mrow >= C) || (sAP[j * C + ar] == 0);

    v8f creg;
#pragma unroll
    for (int r = 0; r < 8; ++r) creg[r] = sqi[r] + sqj;   // |Xi_m|^2 + |Xj_n|^2

    // D = C + A*B = sq_i + sq_j - 2*dot  (full 16x16 squared-distance tile)
    v8f d = __builtin_amdgcn_wmma_f32_16x16x4_f32(
        false, areg, false, breg, (short)0, creg, false, false);

    // Min over atom pairs in the squared domain (sqrt hoisted out).
    float mn = BIGINT_F;
#pragma unroll
    for (int r = 0; r < 8; ++r) {
      bool  pad = (((maski >> (mbase + r)) & 1u) != 0u) || padj;
      float dv  = pad ? BIGINT_F : d[r];
      mn = fminf(mn, dv);
    }
#pragma unroll
    for (int off = 16; off > 0; off >>= 1)
      mn = fminf(mn, __shfl_xor(mn, off, 32));
    if (lane == 0)
      dist[(size_t)wave * L + j] = __builtin_amdgcn_sqrtf(fmaxf(mn, 0.0f));
  }
}

// ---------------------------------------------------------------------------
// Kernel 3: extra_ctx_adj boolean tensor (B,L,L) as 0/1 floats.
// ---------------------------------------------------------------------------
__global__ void adj_kernel(const int* __restrict__ S, const int* __restrict__ seg,
                           float* __restrict__ out_adj, int B, int L) {
  int idx = blockIdx.x * blockDim.x + threadIdx.x;
  if (idx >= B * L * L) return;
  int j = idx % L;
  int t = idx / L;
  int i = t % L;
  int b = t / L;
  int ni = b * L + i, nj = b * L + j;
  int si = S[ni], sj = S[nj];
  bool gi = (si >= 21) && (si <= 23);       // BOA/BOH/BOL
  bool gj = (sj >= 21) && (sj <= 23);
  int  sgi = seg[ni], sgj = seg[nj];
  bool same      = (sgi == sgj);
  bool notself   = (i != j);
  bool notglobal = !(gi || gj);
  bool adjseq    = ((i - j) == 1) || ((j - i) == 1);
  bool global_normal = same && (gi || gj) && notself;
  bool global_global = gi && gj && notself;
  bool seqm          = adjseq && notglobal && (sgi != 1 /*AG_SEG*/);
  out_adj[idx] = (global_normal || global_global || seqm) ? 1.0f : 0.0f;
}

// ---------------------------------------------------------------------------
// Kernel 4: exact k-NN (k=9) over masked distance rows; two passes
// (pass 0 = inner/same-segment "ctx", pass 1 = outer/cross-segment "inter").
// Iterative min-selection with strict < reproduces top_k tie-by-index order.
// ---------------------------------------------------------------------------
__global__ void knn_kernel(const float* __restrict__ dist,
                           const int* __restrict__ S, const int* __restrict__ seg,
                           float* __restrict__ out, int B, int L, int K,
                           size_t oCtxKnn, size_t oCtxValid,
                           size_t oInterKnn, size_t oInterValid) {
  int t = blockIdx.x * blockDim.x + threadIdx.x;
  if (t >= B * L) return;
  int b = t / L;
  int i = t - b * L;
  int ni = t;
  int si = S[ni];
  bool gi = (si >= 21) && (si <= 23);
  int sgi = seg[ni];
  const float* drow = dist + (size_t)t * L;
  int NK = B * L * K;

  for (int pass = 0; pass < 2; ++pass) {
    unsigned long long sel0 = 0ull, sel1 = 0ull;   // 128-bit "already picked"
    size_t oK = (pass == 0) ? oCtxKnn   : oInterKnn;
    size_t oV = (pass == 0) ? oCtxValid : oInterValid;
    for (int kk = 0; kk < K; ++kk) {
      float best = 3.4e38f;
      int bj = 0;
      for (int j = 0; j < L; ++j) {
        bool selected = (j < 64) ? ((sel0 >> j) & 1ull) : ((sel1 >> (j - 64)) & 1ull);
        if (selected) continue;
        int nj = b * L + j;
        int sj = S[nj];
        bool gj   = (sj >= 21) && (sj <= 23);
        bool same = (sgi == seg[nj]);
        bool notg = !(gi || gj);
        bool m = (pass == 0) ? (same && notg && (i != j))
                             : (!same && notg && (i != j));
        float dv = m ? drow[j] : BIGINT_F;
        if (dv < best) { best = dv; bj = j; }
      }
      if (bj < 64) sel0 |= (1ull << bj); else sel1 |= (1ull << (bj - 64));
      bool valid = best < BIGINT_F;
      float src = valid ? (float)ni : -1.0f;
      float dst = valid ? (float)(b * L + bj) : -1.0f;
      out[oK + (size_t)t * K + kk]              = src;   // knn[0] = src
      out[oK + (size_t)NK + (size_t)t * K + kk] = dst;   // knn[1] = dst
      out[oV + (size_t)t * K + kk]              = valid ? 1.0f : 0.0f;
    }
  }
}

// ---------------------------------------------------------------------------
extern "C" void kernel_launch(void* const* d_in, const int* in_sizes, int n_in,
                              void* d_out, int out_size, void* d_ws, size_t ws_size,
                              hipStream_t stream) {
  const int*   S    = (const int*)d_in[0];
  const int*   RP   = (const int*)d_in[1];
  const int*   A    = (const int*)d_in[2];
  const int*   AP   = (const int*)d_in[3];
  const float* X    = (const float*)d_in[4];
  const int*   seg  = (const int*)d_in[5];
  const float* resE = (const float*)d_in[6];
  const float* atmE = (const float*)d_in[7];
  const float* atpE = (const float*)d_in[8];

  int N = in_sizes[0];          // 2048
  int C = in_sizes[2] / N;      // 14
  int E = in_sizes[7] / 16;     // 256 (atom_embed is 16 x E)
  int L = 128;                  // max_n (scalar input; fixed by reference)
  int K = 9;                    // k_neighbors (scalar input; fixed by reference)
  int B = N / L;                // 16

  float* out  = (float*)d_out;
  float* dist = (float*)d_ws;   // B*L*L floats scratch

  // flat output offsets (tuple return order)
  size_t oH         = 0;
  size_t oCtxKnn    = oH + (size_t)N * (size_t)(2 * E);
  size_t NKs        = (size_t)B * L * K;
  size_t oCtxValid  = oCtxKnn + 2 * NKs;
  size_t oAdj       = oCtxValid + NKs;
  size_t oInterKnn  = oAdj + (size_t)B * L * L;
  size_t oInterVal  = oInterKnn + 2 * NKs;

  // 1) embeddings
  embed_kernel<<<N, E, 0, stream>>>(S, RP, A, AP, resE, atmE, atpE, out + oH, C, E);

  // 2) min pairwise atom distances (WMMA): 1 wave per (b,i), 4 waves/block,
  //    all waves of a block in the same batch -> shared LDS staging.
  {
    int waves = B * L;
    int wavesPerBlock = 4;
    int blocks = waves / wavesPerBlock;     // exact cover (L % 4 == 0)
    dist_kernel<<<blocks, wavesPerBlock * 32, 0, stream>>>(X, AP, dist, B, L, C);
  }

  // 3) adjacency tensor
  {
    int total = B * L * L;
    adj_kernel<<<(total + 255) / 256, 256, 0, stream>>>(S, seg, out + oAdj, B, L);
  }

  // 4) dual-mask kNN
  {
    int total = B * L;
    knn_kernel<<<(total + 127) / 128, 128, 0, stream>>>(
        dist, S, seg, out, B, L, K, oCtxKnn, oCtxValid, oInterKnn, oInterVal);
  }
}